// ChaosCatastropheDynamics_43224550867185
// MI455X (gfx1250) — compile-verified
//
#include <hip/hip_runtime.h>
#include <cmath>

typedef float v2f __attribute__((ext_vector_type(2)));
typedef float v8f __attribute__((ext_vector_type(8)));

#define DT_F 0.01f
#define STEPS_N 30

#if __has_builtin(__builtin_amdgcn_tanhf)
__device__ __forceinline__ float fast_tanh(float x) { return __builtin_amdgcn_tanhf(x); }
#else
__device__ __forceinline__ float fast_tanh(float x) { return tanhf(x); }
#endif

__device__ __forceinline__ void lderiv(float sg, float rh, float bt,
                                       float X, float Y, float Z,
                                       float &dx, float &dy, float &dz) {
  dx = sg * (Y - X);
  dy = X * (rh - Z) - Y;
  dz = X * Y - bt * Z;
}

// One wave (32 lanes) processes 32 rows = two 16-row WMMA tiles.
// Phase 1: state0 = x @ W_in^T + b_in       (V_WMMA_F32_16X16X4_F32, K=2048)
// Phase 2: 30 RK4 Lorenz steps + stats      (1 lane per row)
// Phase 3: pert = features @ W_out^T + b_out (WMMA, K=20 zero-padded)
//          out = x + tanh(pert) * |strength|
__global__ __launch_bounds__(128) void chaos_enrich_kernel(
    const float* __restrict__ x, const float* __restrict__ lorenz,
    const float* __restrict__ W_in, const float* __restrict__ b_in,
    const float* __restrict__ W_out, const float* __restrict__ b_out,
    const float* __restrict__ strength, float* __restrict__ out, int D)
{
  constexpr int WAVES = 4;                 // blockDim.x == 128
  __shared__ float lds_c[WAVES][2][16][16]; // C-tile staging (state0 redistribute)
  __shared__ float lds_f[WAVES][32][20];    // features, K padded 18 -> 20 (zeros)

  const int lane = threadIdx.x & 31;
  const int wv   = threadIdx.x >> 5;
  const int half = lane >> 4;   // which 16-lane half
  const int lm   = lane & 15;
  const long long rowBase = ((long long)blockIdx.x * WAVES + wv) * 32;

  // ---------------- Phase 1: state0 via f32 WMMA ----------------
  // A[m][k]: lane holds M=lm, K=2*half+v.  B[k][n]: N=lm, K=2*half+v.
  // B columns n>=3 are zero (W_in has only 3 rows).
  const float wsel = (lm < 3) ? 1.0f : 0.0f;
  const float* __restrict__ wrow = W_in + (size_t)((lm < 3) ? lm : 0) * D;

  for (int t = 0; t < 2; ++t) {
    const float* __restrict__ xrow = x + (size_t)(rowBase + 16 * t + lm) * D;
    v8f c0 = {}; v8f c1 = {};
    for (int kc = 0; kc < D; kc += 8) {     // two K=4 chunks per iter
      const int k0 = kc + 2 * half;         // even -> 8B aligned
      v2f a0 = *(const v2f*)(xrow + k0);
      v2f a1 = *(const v2f*)(xrow + k0 + 4);
      v2f b0 = *(const v2f*)(wrow + k0);
      v2f b1 = *(const v2f*)(wrow + k0 + 4);
      b0 *= wsel; b1 *= wsel;
      c0 = __builtin_amdgcn_wmma_f32_16x16x4_f32(false, a0, false, b0,
                                                 (short)0, c0, false, false);
      c1 = __builtin_amdgcn_wmma_f32_16x16x4_f32(false, a1, false, b1,
                                                 (short)0, c1, false, false);
    }
    c0 = c0 + c1;
    // D layout: lane holds M = 8*half + v, N = lm
#pragma unroll
    for (int v = 0; v < 8; ++v)
      lds_c[wv][t][8 * half + v][lm] = c0[v];
  }
  __syncthreads();

  // ---------------- Phase 2: Lorenz RK4 + stats (lane == row) ----------------
  const float sg = fmaxf(fabsf(lorenz[0]), 0.1f);
  const float rh = fmaxf(fabsf(lorenz[1]), 0.1f);
  const float bt = fmaxf(fabsf(lorenz[2]), 0.1f);

  float X = lds_c[wv][half][lm][0] + b_in[0];
  float Y = lds_c[wv][half][lm][1] + b_in[1];
  float Z = lds_c[wv][half][lm][2] + b_in[2];

  const float fx = X, fy = Y, fz = Z;       // first state
  float sx = 0.f, sy = 0.f, sz = 0.f;       // sums of (s - first)  (shifted moments)
  float qx = 0.f, qy = 0.f, qz = 0.f;       // sums of (s - first)^2
  float mnx = X, mny = Y, mnz = Z;
  float mxx = X, mxy = Y, mxz = Z;

  for (int it = 0; it < STEPS_N; ++it) {
    float k1x,k1y,k1z,k2x,k2y,k2z,k3x,k3y,k3z,k4x,k4y,k4z;
    lderiv(sg,rh,bt, X, Y, Z, k1x,k1y,k1z);
    lderiv(sg,rh,bt, X+0.5f*DT_F*k1x, Y+0.5f*DT_F*k1y, Z+0.5f*DT_F*k1z, k2x,k2y,k2z);
    lderiv(sg,rh,bt, X+0.5f*DT_F*k2x, Y+0.5f*DT_F*k2y, Z+0.5f*DT_F*k2z, k3x,k3y,k3z);
    lderiv(sg,rh,bt, X+DT_F*k3x, Y+DT_F*k3y, Z+DT_F*k3z, k4x,k4y,k4z);
    X += (DT_F/6.0f)*(k1x + 2.f*k2x + 2.f*k3x + k4x);
    Y += (DT_F/6.0f)*(k1y + 2.f*k2y + 2.f*k3y + k4y);
    Z += (DT_F/6.0f)*(k1z + 2.f*k2z + 2.f*k3z + k4z);
    const float dxs = X - fx, dys = Y - fy, dzs = Z - fz;
    sx += dxs; sy += dys; sz += dzs;
    qx += dxs*dxs; qy += dys*dys; qz += dzs*dzs;
    mnx = fminf(mnx, X); mny = fminf(mny, Y); mnz = fminf(mnz, Z);
    mxx = fmaxf(mxx, X); mxy = fmaxf(mxy, Y); mxz = fmaxf(mxz, Z);
  }
  // 31 samples (state0 + 30 steps); shifted-moment mean/std (ddof=1)
  const float inv31 = 1.0f / 31.0f, inv30 = 1.0f / 30.0f;
  const float mex = fx + sx * inv31, mey = fy + sy * inv31, mez = fz + sz * inv31;
  const float vx = fmaxf((qx - sx * sx * inv31) * inv30, 0.f);
  const float vy = fmaxf((qy - sy * sy * inv31) * inv30, 0.f);
  const float vz = fmaxf((qz - sz * sz * inv31) * inv30, 0.f);

  float* f = lds_f[wv][lane];
  f[0] = fx;  f[1] = fy;  f[2] = fz;        // first
  f[3] = X;   f[4] = Y;   f[5] = Z;         // last
  f[6] = mex; f[7] = mey; f[8] = mez;       // mean
  f[9] = sqrtf(vx); f[10] = sqrtf(vy); f[11] = sqrtf(vz); // std
  f[12] = mnx; f[13] = mny; f[14] = mnz;    // min
  f[15] = mxx; f[16] = mxy; f[17] = mxz;    // max
  f[18] = 0.f; f[19] = 0.f;                 // K padding
  __syncthreads();

  // ---------------- Phase 3: perturb GEMM + tanh + residual ----------------
  const float astr = fabsf(strength[0]);
  for (int t = 0; t < 2; ++t) {
    v2f afr[5];
#pragma unroll
    for (int c2 = 0; c2 < 5; ++c2) {
      const int k = 4 * c2 + 2 * half;
      afr[c2].x = lds_f[wv][16 * t + lm][k];
      afr[c2].y = lds_f[wv][16 * t + lm][k + 1];
    }
    const long long rB = rowBase + 16 * t;
    for (int cb = 0; cb < D; cb += 16) {
      const int col = cb + lm;
      const float* __restrict__ wr = W_out + (size_t)col * 18;
      v8f acc = {};
#pragma unroll
      for (int c2 = 0; c2 < 5; ++c2) {
        const int k = 4 * c2 + 2 * half;
        const int   i0 = (k     < 18) ? k       : 0;   // branchless in-bounds
        const int   i1 = (k + 1 < 18) ? (k + 1) : 0;
        const float m0 = (k     < 18) ? 1.f : 0.f;
        const float m1 = (k + 1 < 18) ? 1.f : 0.f;
        v2f b; b.x = wr[i0] * m0; b.y = wr[i1] * m1;
        acc = __builtin_amdgcn_wmma_f32_16x16x4_f32(false, afr[c2], false, b,
                                                    (short)0, acc, false, false);
      }
      const float bo = b_out[col];
#pragma unroll
      for (int v = 0; v < 8; ++v) {
        const long long row = rB + 8 * half + v;
        const size_t idx = (size_t)row * D + col;
        out[idx] = x[idx] + fast_tanh(acc[v] + bo) * astr;
      }
    }
  }
}

extern "C" void kernel_launch(void* const* d_in, const int* in_sizes, int n_in,
                              void* d_out, int out_size, void* d_ws, size_t ws_size,
                              hipStream_t stream) {
  const float* x        = (const float*)d_in[0];
  const float* lorenz   = (const float*)d_in[1];
  const float* W_in     = (const float*)d_in[2];
  const float* b_in     = (const float*)d_in[3];
  const float* W_out    = (const float*)d_in[4];
  const float* b_out    = (const float*)d_in[5];
  const float* strength = (const float*)d_in[6];
  float* out = (float*)d_out;

  const int D = in_sizes[5];                       // b_out length == 2048
  const long long N = (long long)in_sizes[0] / D;  // rows == 32768
  const int rowsPerBlock = 128;                    // 4 waves * 32 rows
  const int grid = (int)((N + rowsPerBlock - 1) / rowsPerBlock);

  chaos_enrich_kernel<<<grid, 128, 0, stream>>>(x, lorenz, W_in, b_in,
                                                W_out, b_out, strength, out, D);
}